// SequenceNLLLoss_72791105732873
// MI455X (gfx1250) — compile-verified
//
#include <hip/hip_runtime.h>

// SequenceNLLLoss for MI455X (gfx1250, wave32).
// B=8, S=512, V=32000 fixed by the reference; B derived from in_sizes[1].
//
// Shape of the computation: scattered gather (<=4096 x 4B) + tiny reductions.
// Latency-bound, zero dense math -> WMMA/TDM intentionally not used; the win
// is one pass over `target` (staged in LDS), non-temporal gathered loads with
// full MLP, and wave32 shuffle reductions.

#define EOS_TOKEN 1
#define S_LEN 512
#define V_LEN 32000
#define NTHREADS 256
#define NWAVES (NTHREADS / 32)

__global__ __launch_bounds__(NTHREADS) void seq_nll_per_seq_kernel(
    const float* __restrict__ logp,    // [B, S, V] log-probs
    const int* __restrict__ target,    // [B, S]
    float* __restrict__ per_seq) {     // [B] output partials (in d_ws)
  const int b   = blockIdx.x;
  const int tid = threadIdx.x;

  __shared__ int   s_tgt[S_LEN];   // stage targets: read HBM once
  __shared__ int   s_eos;
  __shared__ float s_wave[NWAVES];

  if (tid == 0) s_eos = S_LEN;
  __syncthreads();

  // ---- Phase 1: load targets to LDS, find first EOS (order-independent min)
  int local_min = S_LEN;
  const int* tgt_row = target + (size_t)b * S_LEN;
#pragma unroll
  for (int s = tid; s < S_LEN; s += NTHREADS) {
    const int t = tgt_row[s];
    s_tgt[s] = t;
    if (t == EOS_TOKEN && s < local_min) local_min = s;
  }
  if (local_min < S_LEN) atomicMin(&s_eos, local_min);  // ds_min on LDS
  __syncthreads();
  const int eos = s_eos;  // >= 1 by construction of the inputs

  // ---- Phase 2: gather log p(target) for s in [0, eos], streaming (NT) loads.
  // <= 2 independent loads per thread -> all in flight at once.
  const float* seq_logp = logp + (size_t)b * S_LEN * V_LEN;
  float acc = 0.0f;
  for (int s = tid; s <= eos; s += NTHREADS) {
    const float* p = seq_logp + (size_t)s * V_LEN + (size_t)s_tgt[s];
    acc += __builtin_nontemporal_load(p);  // global_load_b32 th:TH_LOAD_NT
  }

  // ---- Phase 3: deterministic reduction. wave32 shuffle tree, then LDS
  // combine across the 8 waves of the block (fixed order every call).
#pragma unroll
  for (int off = 16; off > 0; off >>= 1)
    acc += __shfl_down(acc, off, 32);
  if ((tid & 31) == 0) s_wave[tid >> 5] = acc;
  __syncthreads();
  if (tid == 0) {
    float tot = 0.0f;
#pragma unroll
    for (int w = 0; w < NWAVES; ++w) tot += s_wave[w];
    per_seq[b] = -tot / (float)eos;
  }
}

__global__ void seq_nll_finalize_kernel(const float* __restrict__ per_seq,
                                        float* __restrict__ out, int B) {
  if (blockIdx.x == 0 && threadIdx.x == 0) {
    float s = 0.0f;
    for (int i = 0; i < B; ++i) s += per_seq[i];  // fixed order, deterministic
    out[0] = s;
  }
}

extern "C" void kernel_launch(void* const* d_in, const int* in_sizes, int n_in,
                              void* d_out, int out_size, void* d_ws, size_t ws_size,
                              hipStream_t stream) {
  const float* logp   = (const float*)d_in[0];  // [B*S*V] f32 log-probs
  const int*   target = (const int*)d_in[1];    // [B*S] int32 class ids
  float* per_seq = (float*)d_ws;                // B floats of scratch
  float* out     = (float*)d_out;               // 1 float

  const int B = in_sizes[1] / S_LEN;            // = 8 for the reference shapes

  seq_nll_per_seq_kernel<<<B, NTHREADS, 0, stream>>>(logp, target, per_seq);
  seq_nll_finalize_kernel<<<1, 32, 0, stream>>>(per_seq, out, B);
}